// GraphSAGE_84112639525007
// MI455X (gfx1250) — compile-verified
//
#include <hip/hip_runtime.h>

#define NN 100000      // nodes
#define NE 1600000     // edges
#define D  64          // feature dim

typedef __attribute__((ext_vector_type(2))) float v2f;
typedef __attribute__((ext_vector_type(8))) float v8f;

// ---------------------------------------------------------------- utilities
__global__ void fill_zero_kernel(float* __restrict__ p, int n) {
  int i = blockIdx.x * blockDim.x + threadIdx.x;
  if (i < n) p[i] = 0.0f;
}

// one thread per edge: count in-degree of dst as float
__global__ void degree_kernel(const int* __restrict__ dst, float* __restrict__ cnt) {
  int e = blockIdx.x * blockDim.x + threadIdx.x;
  if (e < NE) atomicAdd(&cnt[dst[e]], 1.0f);
}

// one thread per (edge, 4-float chunk): gather h[src], atomic-accumulate to agg[dst]
// h (25.6MB) and agg (25.6MB) are L2-resident on MI455X (192MB L2) -> gathers hit L2,
// fp32 atomics resolve at L2.
__global__ void scatter_kernel(const float* __restrict__ hin,
                               const int* __restrict__ src,
                               const int* __restrict__ dst,
                               float* __restrict__ agg) {
  long long gid = (long long)blockIdx.x * blockDim.x + threadIdx.x;
  if (gid >= (long long)NE * 16) return;
  int e = (int)(gid >> 4);
  int c = ((int)gid & 15) * 4;
  int s = src[e];
  int d = dst[e];
  const float4 v = *(const float4*)(hin + (long long)s * D + c);
  float* o = agg + (long long)d * D + c;
  atomicAdd(o + 0, v.x);
  atomicAdd(o + 1, v.y);
  atomicAdd(o + 2, v.z);
  atomicAdd(o + 3, v.w);
}

// ---------------------------------------------------------------- node update
// out[16-row tile] = (agg * inv_cnt) @ Wl + h @ Wr + b   (optional ReLU)
// 256 threads = 8 waves; each wave owns one 16-node tile.
// V_WMMA_F32_16X16X4_F32:  D(16x16) = A(16x4) x B(4x16) + C, fp32 throughout.
// A layout (ISA 7.12.2, 32-bit A 16x4): lane L holds row M=L%16;
//   lanes 0-15: v0=K0, v1=K1 ; lanes 16-31: v0=K2, v1=K3.
// B layout (row of B striped across lanes): v0 = row (4k+half), v1 = row (4k+half+2),
//   column = lane%16 (+16*n tile offset).
// C/D layout: VGPR j -> row (j + 8*(lane/16)), col = lane%16.
__global__ __launch_bounds__(256)
void sage_update_wmma(const float* __restrict__ agg,
                      const float* __restrict__ hin,
                      const float* __restrict__ cnt,
                      const float* __restrict__ Wl,
                      const float* __restrict__ Wr,
                      const float* __restrict__ bias,
                      float* __restrict__ out,
                      int relu) {
  __shared__ float sWl[D * D];
  __shared__ float sWr[D * D];
  __shared__ float sB[D];

  for (int i = threadIdx.x; i < D * D; i += 256) {
    sWl[i] = Wl[i];
    sWr[i] = Wr[i];
  }
  if (threadIdx.x < D) sB[threadIdx.x] = bias[threadIdx.x];
  __syncthreads();

  const int wave = threadIdx.x >> 5;
  const int lane = threadIdx.x & 31;
  const int row0 = (blockIdx.x * 8 + wave) * 16;
  if (row0 >= NN) return;   // uniform per wave; no barriers after this point

  const int m    = lane & 15;   // row within tile for A; column-within-16 for B/C
  const int half = lane >> 4;   // K-pair select for A/B fragments

  const int node  = row0 + m;
  const int nodeC = node < NN ? node : NN - 1;   // clamp loads; stores guarded later
  const float* aRow = agg + (long long)nodeC * D;
  const float* hRow = hin + (long long)nodeC * D;
  const float invc  = 1.0f / fmaxf(cnt[nodeC], 1.0f);

  // Preload the 16 K-step A-fragments for both operands (mean-agg and self).
  // Lane holds cols {4k + 2*half, 4k + 2*half + 1} of its row -> 32 floats each.
  float a1[32], a2[32];
#pragma unroll
  for (int k = 0; k < 16; ++k) {
    const int c = 4 * k + 2 * half;
    a1[2 * k]     = aRow[c]     * invc;
    a1[2 * k + 1] = aRow[c + 1] * invc;
    a2[2 * k]     = hRow[c];
    a2[2 * k + 1] = hRow[c + 1];
  }

#pragma unroll
  for (int n = 0; n < 4; ++n) {           // four 16-column output tiles
    const int col = n * 16 + m;
    v8f acc = {};
#pragma unroll
    for (int k = 0; k < 16; ++k) {        // K = 64 in steps of 4
      const int kr = 4 * k + half;        // B row for v0; v1 is kr+2
      v2f av1 = { a1[2 * k], a1[2 * k + 1] };
      v2f av2 = { a2[2 * k], a2[2 * k + 1] };
      v2f bl  = { sWl[kr * D + col], sWl[(kr + 2) * D + col] };
      v2f br  = { sWr[kr * D + col], sWr[(kr + 2) * D + col] };
      acc = __builtin_amdgcn_wmma_f32_16x16x4_f32(false, av1, false, bl,
                                                  (short)0, acc, false, false);
      acc = __builtin_amdgcn_wmma_f32_16x16x4_f32(false, av2, false, br,
                                                  (short)0, acc, false, false);
    }
    const float bcol = sB[col];
#pragma unroll
    for (int j = 0; j < 8; ++j) {
      const int r = row0 + j + half * 8;
      if (r < NN) {                       // per-lane store guard; EXEC reconverges after
        float v = acc[j] + bcol;
        if (relu) v = fmaxf(v, 0.0f);
        out[(long long)r * D + col] = v;
      }
    }
  }
}

// ---------------------------------------------------------------- launch
extern "C" void kernel_launch(void* const* d_in, const int* in_sizes, int n_in,
                              void* d_out, int out_size, void* d_ws, size_t ws_size,
                              hipStream_t stream) {
  const float* x    = (const float*)d_in[0];   // [N, 64]
  const int*   edge = (const int*)d_in[1];     // [2, E]
  const float* Wl   = (const float*)d_in[2];   // [3, 64, 64]
  const float* Wr   = (const float*)d_in[3];   // [3, 64, 64]
  const float* b    = (const float*)d_in[4];   // [3, 64]
  float*       out  = (float*)d_out;           // [N, 64]

  // workspace layout: agg | hA | hB | cnt   (~77 MB total)
  float* agg = (float*)d_ws;
  float* hA  = agg + (size_t)NN * D;
  float* hB  = hA  + (size_t)NN * D;
  float* cnt = hB  + (size_t)NN * D;

  const int* src = edge;
  const int* dst = edge + NE;

  // in-degree (shared by all layers)
  fill_zero_kernel<<<(NN + 255) / 256, 256, 0, stream>>>(cnt, NN);
  degree_kernel<<<(NE + 255) / 256, 256, 0, stream>>>(dst, cnt);

  const float* hin = x;
  float* houts[3] = { hA, hB, out };
  for (int l = 0; l < 3; ++l) {
    fill_zero_kernel<<<(NN * D + 255) / 256, 256, 0, stream>>>(agg, NN * D);
    const long long sthreads = (long long)NE * 16;
    scatter_kernel<<<(int)((sthreads + 255) / 256), 256, 0, stream>>>(hin, src, dst, agg);
    sage_update_wmma<<<(NN + 127) / 128, 256, 0, stream>>>(
        agg, hin, cnt, Wl + l * D * D, Wr + l * D * D, b + l * D,
        houts[l], (l < 2) ? 1 : 0);
    hin = houts[l];
  }
}